// TalkingHeadAttention_34179349741908
// MI455X (gfx1250) — compile-verified
//
#include <hip/hip_runtime.h>

// Problem constants
constexpr int kB  = 8;
constexpr int kN  = 1024;
constexpr int kC  = 512;
constexpr int kH  = 8;
constexpr int kDH = 64;
#define SCALE_F 0.125f      // DH^-0.5
#define NEG_F  (-1.0e9f)

typedef __attribute__((ext_vector_type(2))) float v2f;
typedef __attribute__((ext_vector_type(8))) float v8f;

#ifndef __has_builtin
#define __has_builtin(x) 0
#endif
#if __has_builtin(__builtin_amdgcn_global_load_async_to_lds_b32)
#define HAVE_ASYNC_LDS 1
#else
#define HAVE_ASYNC_LDS 0
#endif

#if defined(__HIP_DEVICE_COMPILE__)
typedef __attribute__((address_space(1))) int as1_int;
typedef __attribute__((address_space(3))) int as3_int;

// Generic -> AS1 (global): same 64-bit address value on AMDGPU.
__device__ __forceinline__ as1_int* to_global_as(const void* p) {
  return (as1_int*)(unsigned long long)p;
}
// Generic LDS flat pointer -> AS3 offset: low 32 bits are the LDS offset
// (ISA aperture rule).
__device__ __forceinline__ as3_int* to_lds_as(void* p) {
  return (as3_int*)(unsigned int)(unsigned long long)p;
}
#endif

// Wait for all of this wave's outstanding async LDS loads (ASYNCcnt == 0).
__device__ __forceinline__ void wait_async_all() {
#if HAVE_ASYNC_LDS
#if __has_builtin(__builtin_amdgcn_s_wait_asynccnt)
  __builtin_amdgcn_s_wait_asynccnt(0);
#else
  asm volatile("s_wait_asynccnt 0x0" ::: "memory");
#endif
#endif
}

// ---------------------------------------------------------------------------
// One 16x16 fp32 output tile via V_WMMA_F32_16X16X4_F32, accumulating over K.
// A  : row-major [16 x K], leading dim lda   (element A[m*lda + k])
// Bt : row-major [16 x K], leading dim ldb   (out[m][n] = sum_k A[m][k]*Bt[n][k])
// Per ISA layout for 32-bit 16x4 A: lane L holds rows m=L%16, K pair 2*(L/16).
// ---------------------------------------------------------------------------
__device__ __forceinline__ v8f wmma_tile_f32(const float* __restrict__ A, int lda,
                                             const float* __restrict__ Bt, int ldb,
                                             int K, v8f acc) {
  const int lane = threadIdx.x & 31;
  const int mr   = lane & 15;
  const int kq   = (lane >> 4) << 1;
  const float* ap = A  + mr * lda + kq;
  const float* bp = Bt + mr * ldb + kq;
  for (int k = 0; k < K; k += 4) {
    v2f a = *(const v2f*)(ap + k);
    v2f b = *(const v2f*)(bp + k);
    acc = __builtin_amdgcn_wmma_f32_16x16x4_f32(false, a, false, b,
                                                (short)0, acc, false, false);
  }
  return acc;
}

// ---------------------------------------------------------------------------
// K1: qkv = x @ qkv_w^T + qkv_b, scattered to [3][B][H][N][DH], q pre-scaled.
// grid: (8192/16)*(1536/16)/4 blocks of 128 threads (4 waves, 1 tile each).
// ---------------------------------------------------------------------------
__global__ void k_qkv(const float* __restrict__ x, const float* __restrict__ w,
                      const float* __restrict__ bias, float* __restrict__ qkv) {
  const int wid  = threadIdx.x >> 5;
  const int lane = threadIdx.x & 31;
  const int tile = blockIdx.x * 4 + wid;
  const int NT   = (3 * kC) / 16;          // 96 column tiles
  const int mt   = tile / NT;
  const int nt   = tile % NT;
  v8f acc = {};
  acc = wmma_tile_f32(x + (size_t)mt * 16 * kC, kC,
                      w + (size_t)nt * 16 * kC, kC, kC, acc);
  const int col   = nt * 16 + (lane & 15);
  const int which = col / kC;              // 0=q 1=k 2=v
  const int h     = (col % kC) / kDH;
  const int dh    = col % kDH;
  const float bv  = bias[col];
  const float scl = (which == 0) ? SCALE_F : 1.0f;
#pragma unroll
  for (int r = 0; r < 8; ++r) {
    const int row = mt * 16 + r + ((lane >> 4) << 3);
    const int b   = row / kN;
    const int n   = row % kN;
    qkv[(((size_t)(which * kB + b) * kH + h) * kN + n) * kDH + dh] =
        (acc[r] + bv) * scl;
  }
}

// ---------------------------------------------------------------------------
// K2: per (b, n-tile, m-tile): wave h computes S_h = q_h @ k_h^T (16x16),
// mix all 8 heads via LDS with pl_w, add pl_b and key mask, write pre-softmax
// logits into the attn output region (scratch use; overwritten by K3).
// grid: (64,64,8), 256 threads (8 waves).
// ---------------------------------------------------------------------------
__global__ void k_scores(const float* __restrict__ qkv,
                         const float* __restrict__ pl_w,
                         const float* __restrict__ pl_b,
                         const float* __restrict__ mask,
                         float* __restrict__ attn) {
  __shared__ float s[kH][16][17];
  const int h    = threadIdx.x >> 5;
  const int lane = threadIdx.x & 31;
  const int b    = blockIdx.z;
  const int nt   = blockIdx.y;
  const int mt   = blockIdx.x;
  const float* qh = qkv + (((size_t)(0 * kB + b) * kH + h) * kN + nt * 16) * kDH;
  const float* kh = qkv + (((size_t)(1 * kB + b) * kH + h) * kN + mt * 16) * kDH;
  v8f acc = {};
  acc = wmma_tile_f32(qh, kDH, kh, kDH, kDH, acc);
#pragma unroll
  for (int r = 0; r < 8; ++r)
    s[h][r + ((lane >> 4) << 3)][lane & 15] = acc[r];
  __syncthreads();
  // wave h now produces output head g = h
  const int g = h;
  const int m = mt * 16 + (lane & 15);
  const float base = pl_b[g] + (1.0f - mask[b * kN + m]) * NEG_F;
  float wr[kH];
#pragma unroll
  for (int j = 0; j < kH; ++j) wr[j] = pl_w[g * kH + j];
#pragma unroll
  for (int r = 0; r < 8; ++r) {
    const int row = r + ((lane >> 4) << 3);
    float v = base;
#pragma unroll
    for (int j = 0; j < kH; ++j) v += s[j][row][lane & 15] * wr[j];
    attn[((size_t)(b * kH + g) * kN + nt * 16 + row) * kN + m] = v;
  }
}

// ---------------------------------------------------------------------------
// K3: fused row softmax + post-softmax talking-heads mix, IN PLACE on attn.
// One block per (b,n): wave h softmaxes its head's 1024-key row (wave32
// shuffle reductions), normalized P kept in 32KB LDS, then each wave g mixes
// across heads with pw_w/pw_b and writes the final attn back.
// grid: (1024, 8), 256 threads.
// ---------------------------------------------------------------------------
__global__ void k_softmax_mix(float* __restrict__ attn,
                              const float* __restrict__ pw_w,
                              const float* __restrict__ pw_b) {
  __shared__ float rows[kH][kN];           // 32 KB
  const int h    = threadIdx.x >> 5;
  const int lane = threadIdx.x & 31;
  const int b    = blockIdx.y;
  const int n    = blockIdx.x;
  float* rowp = attn + ((size_t)(b * kH + h) * kN + n) * kN;
  float vals[32];
  float mx = -3.0e38f;
#pragma unroll
  for (int i = 0; i < 32; ++i) {
    vals[i] = rowp[lane + i * 32];
    mx = fmaxf(mx, vals[i]);
  }
#pragma unroll
  for (int off = 16; off > 0; off >>= 1)
    mx = fmaxf(mx, __shfl_xor(mx, off, 32));
  float sum = 0.0f;
#pragma unroll
  for (int i = 0; i < 32; ++i) {
    vals[i] = __expf(vals[i] - mx);
    sum += vals[i];
  }
#pragma unroll
  for (int off = 16; off > 0; off >>= 1)
    sum += __shfl_xor(sum, off, 32);
  const float inv = 1.0f / sum;
#pragma unroll
  for (int i = 0; i < 32; ++i)
    rows[h][lane + i * 32] = vals[i] * inv;
  __syncthreads();
  // wave h writes output head g = h
  const int g = h;
  float wg[kH];
#pragma unroll
  for (int j = 0; j < kH; ++j) wg[j] = pw_w[g * kH + j];
  const float bg = pw_b[g];
  float* outp = attn + ((size_t)(b * kH + g) * kN + n) * kN;
  for (int m = lane; m < kN; m += 32) {
    float o = bg;
#pragma unroll
    for (int j = 0; j < kH; ++j) o += wg[j] * rows[j][m];
    outp[m] = o;
  }
}

// ---------------------------------------------------------------------------
// Stage one 64-key x 64-dim V chunk, transposed, into LDS buffer vt[d][m].
// Async path: GLOBAL_LOAD_ASYNC_TO_LDS_B32 (per-lane LDS destination address
// performs the transpose in flight; tracked by ASYNCcnt). Fallback: regular
// load + ds_store.
// ---------------------------------------------------------------------------
__device__ __forceinline__ void load_v_chunk(const float* __restrict__ vh,
                                             float (*__restrict__ vt)[68],
                                             int mc, int tid) {
  for (int idx = tid; idx < 64 * 64; idx += 128) {
    const int mi = idx >> 6;
    const int d  = idx & 63;
    const float* src = vh + (size_t)(mc + mi) * kDH + d;
#if HAVE_ASYNC_LDS && defined(__HIP_DEVICE_COMPILE__)
    __builtin_amdgcn_global_load_async_to_lds_b32(
        to_global_as(src), to_lds_as(&vt[d][mi]), 0, 0);
#else
    vt[d][mi] = *src;
#endif
  }
}

// ---------------------------------------------------------------------------
// K4: out_heads[b][g][n][d] = attn[b][g][n][:] @ v[b][g][:][d]  (WMMA f32).
// V chunks double-buffered in LDS via async-to-LDS copies (transposed layout,
// stride 68: 8B-aligned ds_load_b64, conflict-free banks); next attn A-tile
// prefetched (global_prefetch_b8) while the current chunk's WMMAs run.
// 4 waves cover the 4 d-tiles; merged store to tmp [B][N][C].
// grid: (64, 8, 8), 128 threads.
// ---------------------------------------------------------------------------
__global__ void k_av(const float* __restrict__ attn, const float* __restrict__ qkv,
                     float* __restrict__ tmp) {
  __shared__ float vt[2][kDH][68];         // ~34.8 KB
  const int w    = threadIdx.x >> 5;
  const int lane = threadIdx.x & 31;
  const int tid  = threadIdx.x;
  const int b    = blockIdx.z;
  const int g    = blockIdx.y;
  const int nt   = blockIdx.x;
  const float* vh   = qkv + (((size_t)(2 * kB + b) * kH + g) * kN) * kDH;
  const float* arow = attn + ((size_t)(b * kH + g) * kN + nt * 16) * kN;
  const int mr = lane & 15;
  const int kq = (lane >> 4) << 1;
  v8f acc = {};

  load_v_chunk(vh, vt[0], 0, tid);
  wait_async_all();
  __syncthreads();

  for (int c = 0; c < kN / 64; ++c) {
    const int mc = c * 64;
    if (c + 1 < kN / 64)
      load_v_chunk(vh, vt[(c + 1) & 1], mc + 64, tid);   // overlap with compute

    const float* ap = arow + (size_t)mr * kN + mc + kq;
    const float* bp = &vt[c & 1][w * 16 + mr][kq];
    if (c + 1 < kN / 64)
      __builtin_prefetch(ap + 64, 0, 1);                 // next A chunk -> cache
#pragma unroll
    for (int k = 0; k < 64; k += 4) {
      v2f a  = *(const v2f*)(ap + k);
      v2f bb = *(const v2f*)(bp + k);
      acc = __builtin_amdgcn_wmma_f32_16x16x4_f32(false, a, false, bb,
                                                  (short)0, acc, false, false);
    }
    wait_async_all();      // next chunk's async LDS writes landed
    __syncthreads();       // publish to all waves / protect buffer reuse
  }
#pragma unroll
  for (int r = 0; r < 8; ++r) {
    const int row = nt * 16 + r + ((lane >> 4) << 3);
    tmp[((size_t)b * kN + row) * kC + g * kDH + w * 16 + (lane & 15)] = acc[r];
  }
}

// ---------------------------------------------------------------------------
// K5: out = tmp @ proj_w^T + proj_b. grid: (8192/16)*(512/16)/4 blocks x 128.
// ---------------------------------------------------------------------------
__global__ void k_proj(const float* __restrict__ tmp, const float* __restrict__ w,
                       const float* __restrict__ bias, float* __restrict__ out) {
  const int wid  = threadIdx.x >> 5;
  const int lane = threadIdx.x & 31;
  const int tile = blockIdx.x * 4 + wid;
  const int NT   = kC / 16;                // 32
  const int mt   = tile / NT;
  const int nt   = tile % NT;
  v8f acc = {};
  acc = wmma_tile_f32(tmp + (size_t)mt * 16 * kC, kC,
                      w + (size_t)nt * 16 * kC, kC, kC, acc);
  const int col  = nt * 16 + (lane & 15);
  const float bv = bias[col];
#pragma unroll
  for (int r = 0; r < 8; ++r) {
    const int row = mt * 16 + r + ((lane >> 4) << 3);
    out[(size_t)row * kC + col] = acc[r] + bv;
  }
}

// ---------------------------------------------------------------------------
extern "C" void kernel_launch(void* const* d_in, const int* in_sizes, int n_in,
                              void* d_out, int out_size, void* d_ws, size_t ws_size,
                              hipStream_t stream) {
  (void)in_sizes; (void)n_in; (void)out_size; (void)ws_size;
  const float* x      = (const float*)d_in[0];
  const float* mask   = (const float*)d_in[1];
  const float* qkv_w  = (const float*)d_in[2];
  const float* qkv_b  = (const float*)d_in[3];
  const float* proj_w = (const float*)d_in[4];
  const float* proj_b = (const float*)d_in[5];
  const float* pl_w   = (const float*)d_in[6];
  const float* pl_b   = (const float*)d_in[7];
  const float* pw_w   = (const float*)d_in[8];
  const float* pw_b   = (const float*)d_in[9];

  float* out  = (float*)d_out;                              // [B,N,C]
  float* attn = out + (size_t)kB * kN * kC;                 // [B,H,N,N]
  float* qkv  = (float*)d_ws;                               // [3,B,H,N,DH]  50.3 MB
  float* tmp  = qkv + (size_t)3 * kB * kH * kN * kDH;       // [B,N,C]       16.8 MB

  // 1) QKV projection (WMMA f32), q pre-scaled
  k_qkv<<<dim3((512 * 96) / 4), 128, 0, stream>>>(x, qkv_w, qkv_b, qkv);
  // 2) scores + pre-softmax head mix + mask -> attn region (scratch)
  k_scores<<<dim3(64, 64, 8), 256, 0, stream>>>(qkv, pl_w, pl_b, mask, attn);
  // 3) fused softmax + post-softmax head mix, in place
  k_softmax_mix<<<dim3(kN, kB), 256, 0, stream>>>(attn, pw_w, pw_b);
  // 4) attn @ v -> tmp (head-merged), async double-buffered V staging
  k_av<<<dim3(64, 8, 8), 128, 0, stream>>>(attn, qkv, tmp);
  // 5) output projection
  k_proj<<<dim3((512 * 32) / 4), 128, 0, stream>>>(tmp, proj_w, proj_b, out);
}